// TrajectoryScore_58145267253396
// MI455X (gfx1250) — compile-verified
//
#include <hip/hip_runtime.h>

// Precision-sensitive streaming reduction -> keep everything f32.
// Memory-bound: 153.6 MB in / 23.3 TB/s ~= 6.6 us floor. Use b128 loads
// (each thread owns 4 obs = 48B = 3x float4, 16B aligned), fast TRANS
// exp/log, and a WMMA-based wave reduction (deterministic, no atomics).

typedef float v2f __attribute__((ext_vector_type(2)));
typedef float v8f __attribute__((ext_vector_type(8)));

#if defined(__has_builtin)
#if __has_builtin(__builtin_amdgcn_wmma_f32_16x16x4_f32)
#define TS_HAS_WMMA_F32X4 1
#endif
#endif

// Sum x over all 32 lanes of the wave; every lane returns the total.
// WMMA path: A = {x, 0} per lane -> A(16x4) rowsum(m) = x(lane m) + x(lane m+16).
// B = all ones (layout-irrelevant). Per documented C/D layout, summing the 8
// D VGPRs per lane yields sum(rows 0-7) in lanes 0-15 and sum(rows 8-15) in
// lanes 16-31; shfl_xor(16) combines them. EXEC must be all ones (callers
// invoke from converged full blocks).
__device__ __forceinline__ float wave_sum32(float x) {
#ifdef TS_HAS_WMMA_F32X4
  v2f a; a[0] = x; a[1] = 0.0f;
  v2f b; b[0] = 1.0f; b[1] = 1.0f;
  v8f c = {0.f, 0.f, 0.f, 0.f, 0.f, 0.f, 0.f, 0.f};
  v8f d = __builtin_amdgcn_wmma_f32_16x16x4_f32(
      /*neg_a=*/false, a, /*neg_b=*/false, b,
      /*c_mod=*/(short)0, c, /*reuse_a=*/false, /*reuse_b=*/false);
  float t = ((d[0] + d[1]) + (d[2] + d[3])) + ((d[4] + d[5]) + (d[6] + d[7]));
  t += __shfl_xor(t, 16, 32);
  return t;
#else
  for (int off = 16; off > 0; off >>= 1) x += __shfl_xor(x, off, 32);
  return x;
#endif
}

__global__ void __launch_bounds__(256) ts_main_kernel(
    const float* __restrict__ u_pred, const float* __restrict__ u_obs,
    const float* __restrict__ H, const float* __restrict__ LAM,
    const float* __restrict__ TH, float* __restrict__ ws, int nper) {
  const int seg = blockIdx.y;
  const int groups = nper >> 2;  // 4 observations per thread-group

  // Per-segment constants (uniform across the block).
  const float h      = H[seg];
  const float lam    = LAM[seg];
  const float th     = TH[seg];
  const float inv_th = 1.0f / th;
  const float c1     = h * lam / (1.0f - __expf(-lam));  // p_hit = c1 * e
  const float c2     = 1.0f - h;                          // p = p_hit + c2

  const float* pseg = u_pred + (long long)seg * nper * 3;
  const float* oseg = u_obs  + (long long)seg * nper * 3;

  float acc_ll = 0.0f, acc_hh = 0.0f;

  const int stride = gridDim.x * blockDim.x;
  for (int g = blockIdx.x * blockDim.x + threadIdx.x; g < groups; g += stride) {
    // 4 obs = 12 floats = 3 x float4, byte offset 48*g (16B aligned).
    const float4* P = reinterpret_cast<const float4*>(pseg + 12ll * g);
    const float4* O = reinterpret_cast<const float4*>(oseg + 12ll * g);
    float4 p0 = P[0], p1 = P[1], p2 = P[2];
    float4 o0 = O[0], o1 = O[1], o2 = O[2];

    float dx[4], dy[4], dz[4];
    dx[0] = p0.x - o0.x; dy[0] = p0.y - o0.y; dz[0] = p0.z - o0.z;
    dx[1] = p0.w - o0.w; dy[1] = p1.x - o1.x; dz[1] = p1.y - o1.y;
    dx[2] = p1.z - o1.z; dy[2] = p1.w - o1.w; dz[2] = p2.x - o2.x;
    dx[3] = p2.y - o2.y; dy[3] = p2.z - o2.z; dz[3] = p2.w - o2.w;

#pragma unroll
    for (int k = 0; k < 4; ++k) {
      float s2   = dx[k] * dx[k] + dy[k] * dy[k] + dz[k] * dz[k];
      float v    = s2 * inv_th;
      float e    = __expf(-lam * v);        // v_exp_f32 (TRANS)
      float ph   = c1 * e;                  // h * emlx * lam / (1 - exp(-lam))
      float p    = ph + c2;                 // p_hit + (1 - h)  (> 0 always)
      float lp   = __logf(p);               // v_log_f32 (TRANS)
      float post = __fdividef(ph, p);       // p_hit_post
      bool close = s2 < th;
      acc_ll += close ? lp : 0.0f;
      acc_hh += (close && (post > 0.95f)) ? post : 0.0f;
    }
  }

  // Converged here: full 8-wave block, EXEC all ones -> WMMA-legal.
  float w_ll = wave_sum32(acc_ll);
  float w_hh = wave_sum32(acc_hh);

  __shared__ float red[16];  // 8 waves x 2 values
  const int lane = threadIdx.x & 31;
  const int wv   = threadIdx.x >> 5;
  if (lane == 0) { red[2 * wv] = w_ll; red[2 * wv + 1] = w_hh; }
  __syncthreads();
  if (threadIdx.x == 0) {
    float s0 = 0.0f, s1 = 0.0f;
#pragma unroll
    for (int w = 0; w < 8; ++w) { s0 += red[2 * w]; s1 += red[2 * w + 1]; }
    float* o = ws + 2ll * ((long long)seg * gridDim.x + blockIdx.x);
    o[0] = s0;
    o[1] = s1;
  }
}

// One thread per segment: deterministic fixed-order sum of the per-block
// partials; writes all 192 outputs (log_like, hits, hits_raw = hits).
__global__ void ts_finalize_kernel(const float* __restrict__ ws,
                                   float* __restrict__ out, int bps, int B) {
  int i = blockIdx.x * blockDim.x + threadIdx.x;
  if (i < B) {
    const float* w = ws + 2ll * (long long)i * bps;
    float a = 0.0f, b = 0.0f;
    for (int k = 0; k < bps; ++k) { a += w[2 * k]; b += w[2 * k + 1]; }
    out[i]         = a;  // log_like
    out[B + i]     = b;  // hits
    out[2 * B + i] = b;  // hits_raw
  }
}

extern "C" void kernel_launch(void* const* d_in, const int* in_sizes, int n_in,
                              void* d_out, int out_size, void* d_ws, size_t ws_size,
                              hipStream_t stream) {
  const float* u_pred = (const float*)d_in[0];
  const float* u_obs  = (const float*)d_in[1];
  const float* h      = (const float*)d_in[2];
  const float* lam    = (const float*)d_in[3];
  const float* th     = (const float*)d_in[4];
  float* out = (float*)d_out;
  float* ws  = (float*)d_ws;

  const int B    = in_sizes[2];                 // 64 segments
  const int N    = in_sizes[0] / 3;             // 6,400,000 observations
  const int nper = N / B;                       // 100,000 (multiple of 4)

  int bps = 32;                                  // blocks per segment
  size_t need = (size_t)B * bps * 2 * sizeof(float);
  if (ws_size < need) {
    bps = (int)(ws_size / ((size_t)B * 2 * sizeof(float)));
    if (bps < 1) bps = 1;
  }

  dim3 grid(bps, B);
  ts_main_kernel<<<grid, 256, 0, stream>>>(u_pred, u_obs, h, lam, th, ws, nper);
  ts_finalize_kernel<<<1, ((B + 63) / 64) * 64, 0, stream>>>(ws, out, bps, B);
}